// AutoFocalLoss_Regression_59957743452451
// MI455X (gfx1250) — compile-verified
//
#include <hip/hip_runtime.h>
#include <math.h>

// ---------------------------------------------------------------------------
// AutoFocalLoss_Regression — two-pass deterministic reduction for MI455X.
//
//   loss = mean(d) * mean_erf^gamma + log(var+1),
//   gamma = -log(1 - mean_erf),   mean_erf = mean(erf(d * INV_SQRT2 / var))
//
// Pass 1: sum(d), sum(d^2)     -> mean, var   (block partials, no atomics)
// Pass 2: sum(erf(d*s))        -> mean_erf    (re-reads inputs; 128 MiB
//                                              working set fits 192 MB L2)
// Cross-lane sums use V_WMMA_F32_16X16X4_F32 with an all-ones B matrix
// (layout of B irrelevant since every element is 1; A/D layouts per ISA).
// ---------------------------------------------------------------------------

typedef float v2f __attribute__((ext_vector_type(2)));
typedef float v8f __attribute__((ext_vector_type(8)));

#define RED_BLOCK 256
#define RED_GRID  2048
#define INV_SQRT2_F 0.70710678118654752440f

// Sum v across all 32 lanes of the wave (result broadcast to every lane).
// A = {v, 0} per lane => A[m][0]=v_m (lanes 0-15), A[m][2]=v_{m+16}; B = ones
// => D[m][n] = v_m + v_{m+16}. Lane l holds D rows (l<16 ? 0..7 : 8..15) in
// its 8 C VGPRs; summing them + shfl_xor(16) yields the full 32-lane sum.
__device__ __forceinline__ float wave_sum32(float v) {
    v2f a; a.x = v;    a.y = 0.0f;
    v2f b; b.x = 1.0f; b.y = 1.0f;
    v8f c = {0.f, 0.f, 0.f, 0.f, 0.f, 0.f, 0.f, 0.f};
    c = __builtin_amdgcn_wmma_f32_16x16x4_f32(
            /*neg_a=*/false, a, /*neg_b=*/false, b,
            /*c_mod=*/(short)0, c, /*reuse_a=*/false, /*reuse_b=*/false);
    float s = ((c[0] + c[1]) + (c[2] + c[3])) + ((c[4] + c[5]) + (c[6] + c[7]));
    s += __shfl_xor(s, 16, 32);
    return s;
}

// Block-wide (256 threads = 8 wave32) sum; result broadcast to all threads.
__device__ __forceinline__ float block_sum256(float v) {
    __shared__ float lds[8];
    float w = wave_sum32(v);
    unsigned lane = threadIdx.x & 31u;
    unsigned wid  = threadIdx.x >> 5;
    __syncthreads();                 // guard lds reuse across successive calls
    if (lane == 0) lds[wid] = w;
    __syncthreads();
    float t = (lane < 8u) ? lds[lane] : 0.0f;
    return wave_sum32(t);            // uniform full-wave exec (EXEC all ones)
}

// ---------------------------------------------------------------------------
// Pass 1: per-block partial {sum|d|, sum d^2}
// ---------------------------------------------------------------------------
__global__ void __launch_bounds__(RED_BLOCK)
pass1_kernel(const float* __restrict__ pred, const float* __restrict__ tgt,
             float2* __restrict__ partials, size_t n) {
    const size_t n4     = n >> 2;
    const size_t tid    = (size_t)blockIdx.x * blockDim.x + threadIdx.x;
    const size_t stride = (size_t)gridDim.x * blockDim.x;
    const float4* __restrict__ p4 = (const float4*)pred;
    const float4* __restrict__ t4 = (const float4*)tgt;

    float sd = 0.0f, sq = 0.0f;
    for (size_t i = tid; i < n4; i += stride) {
        if (i + stride < n4) {                       // gfx1250 global_prefetch_b8
            __builtin_prefetch(p4 + i + stride, 0, 1);
            __builtin_prefetch(t4 + i + stride, 0, 1);
        }
        float4 p = p4[i];
        float4 t = t4[i];
        float d0 = fabsf(p.x - t.x);
        float d1 = fabsf(p.y - t.y);
        float d2 = fabsf(p.z - t.z);
        float d3 = fabsf(p.w - t.w);
        sd += (d0 + d1) + (d2 + d3);
        sq += (d0 * d0 + d1 * d1) + (d2 * d2 + d3 * d3);
    }
    if (blockIdx.x == 0) {                           // scalar tail (n % 4)
        for (size_t i = (n4 << 2) + threadIdx.x; i < n; i += blockDim.x) {
            float d = fabsf(pred[i] - tgt[i]);
            sd += d;
            sq += d * d;
        }
    }
    float bsd = block_sum256(sd);
    float bsq = block_sum256(sq);
    if (threadIdx.x == 0) partials[blockIdx.x] = make_float2(bsd, bsq);
}

// ---------------------------------------------------------------------------
// Reduce pass-1 partials -> mean, var, erf-scale (single block, deterministic)
// ---------------------------------------------------------------------------
__global__ void __launch_bounds__(RED_BLOCK)
reduce1_kernel(const float2* __restrict__ partials, int m,
               float* __restrict__ scalars, unsigned long long n) {
    float sd = 0.0f, sq = 0.0f;
    for (int i = threadIdx.x; i < m; i += RED_BLOCK) {
        float2 v = partials[i];
        sd += v.x;
        sq += v.y;
    }
    float Sd = block_sum256(sd);
    float Sq = block_sum256(sq);
    if (threadIdx.x == 0) {
        float nn   = (float)n;
        float mean = Sd / nn;
        float var  = (Sq - Sd * mean) / (nn - 1.0f);   // unbiased (ddof=1)
        scalars[0] = mean;
        scalars[1] = var;
        scalars[2] = INV_SQRT2_F / var;                 // erf argument scale
    }
}

// ---------------------------------------------------------------------------
// Pass 2: per-block partial sum of erf(d * scale); inputs should hit L2
// ---------------------------------------------------------------------------
__global__ void __launch_bounds__(RED_BLOCK)
pass2_kernel(const float* __restrict__ pred, const float* __restrict__ tgt,
             const float* __restrict__ scalars, float* __restrict__ partials2,
             size_t n) {
    const float  s      = scalars[2];
    const size_t n4     = n >> 2;
    const size_t tid    = (size_t)blockIdx.x * blockDim.x + threadIdx.x;
    const size_t stride = (size_t)gridDim.x * blockDim.x;
    const float4* __restrict__ p4 = (const float4*)pred;
    const float4* __restrict__ t4 = (const float4*)tgt;

    float acc = 0.0f;
    for (size_t i = tid; i < n4; i += stride) {
        if (i + stride < n4) {
            __builtin_prefetch(p4 + i + stride, 0, 1);
            __builtin_prefetch(t4 + i + stride, 0, 1);
        }
        float4 p = p4[i];
        float4 t = t4[i];
        float e0 = erff(fabsf(p.x - t.x) * s);
        float e1 = erff(fabsf(p.y - t.y) * s);
        float e2 = erff(fabsf(p.z - t.z) * s);
        float e3 = erff(fabsf(p.w - t.w) * s);
        acc += (e0 + e1) + (e2 + e3);
    }
    if (blockIdx.x == 0) {
        for (size_t i = (n4 << 2) + threadIdx.x; i < n; i += blockDim.x) {
            acc += erff(fabsf(pred[i] - tgt[i]) * s);
        }
    }
    float tot = block_sum256(acc);
    if (threadIdx.x == 0) partials2[blockIdx.x] = tot;
}

// ---------------------------------------------------------------------------
// Final: mean_erf -> gamma -> scalar loss
// ---------------------------------------------------------------------------
__global__ void __launch_bounds__(RED_BLOCK)
final_kernel(const float* __restrict__ partials2, int m,
             const float* __restrict__ scalars, float* __restrict__ out,
             unsigned long long n) {
    float se = 0.0f;
    for (int i = threadIdx.x; i < m; i += RED_BLOCK) se += partials2[i];
    float Se = block_sum256(se);
    if (threadIdx.x == 0) {
        float mean_d   = scalars[0];
        float var      = scalars[1];
        float mean_erf = Se / (float)n;                // == 1 - p_correct
        float p        = 1.0f - mean_erf;              // p_correct
        float gamma    = -logf(p);
        // mean(loss) = mean(d) * (1-p)^gamma + log(var+1)   (scalar factor!)
        out[0] = mean_d * powf(mean_erf, gamma) + logf(var + 1.0f);
    }
}

// ---------------------------------------------------------------------------
extern "C" void kernel_launch(void* const* d_in, const int* in_sizes, int n_in,
                              void* d_out, int out_size, void* d_ws, size_t ws_size,
                              hipStream_t stream) {
    const float* pred = (const float*)d_in[0];
    const float* tgt  = (const float*)d_in[1];
    float*       out  = (float*)d_out;
    const size_t n    = (size_t)in_sizes[0];

    // Workspace layout (all rewritten every call; ~25 KB total):
    float2* partials1 = (float2*)d_ws;                                   // 16 KB
    float*  scalars   = (float*)((char*)d_ws + RED_GRID * sizeof(float2));
    float*  partials2 = (float*)((char*)d_ws + RED_GRID * sizeof(float2) + 256);

    pass1_kernel  <<<RED_GRID, RED_BLOCK, 0, stream>>>(pred, tgt, partials1, n);
    reduce1_kernel<<<1,        RED_BLOCK, 0, stream>>>(partials1, RED_GRID, scalars,
                                                       (unsigned long long)n);
    pass2_kernel  <<<RED_GRID, RED_BLOCK, 0, stream>>>(pred, tgt, scalars, partials2, n);
    final_kernel  <<<1,        RED_BLOCK, 0, stream>>>(partials2, RED_GRID, scalars, out,
                                                       (unsigned long long)n);
}